// IntraSampleNTXEntLoss_52484500357142
// MI455X (gfx1250) — compile-verified
//
#include <hip/hip_runtime.h>

// Problem constants (from reference): views (S=4, V=2, B=8192, D=512) f32
#define S_   4
#define V_   2
#define B_   8192
#define D_   512
#define INV_TEMP 10.0f          // 1 / 0.1

typedef __attribute__((ext_vector_type(2))) float v2f;
typedef __attribute__((ext_vector_type(8))) float v8f;

// One wave handles 2 batch elements: Z = 16 x 512 tile (rows = (s*2+v) for b0,
// then for b1).  G = Z * Z^T via chained V_WMMA_F32_16X16X4_F32.
// A(16x4) and B(4x16) f32 operands have identical register images for Z/Z^T,
// so we feed the same registers to both -> Gram; intra-group-of-4 k order is
// irrelevant (sum over all k).  Two independent accumulators break the WMMA
// RAW chain; unroll-8 keeps 8 b128 loads in flight per wave for MLP.
__global__ __launch_bounds__(256) void ntxent_gram_kernel(
    const float* __restrict__ views, float* __restrict__ partial)
{
    __shared__ float gtile[8][256];   // per-wave 16x16 Gram dump
    __shared__ float lred[8][8];      // per-wave loss partials
    __shared__ float wsum[8];

    const int tid  = threadIdx.x;
    const int wave = tid >> 5;
    const int lane = tid & 31;
    const int tile = blockIdx.x * 8 + wave;     // 0 .. B/2-1 (4096 tiles)
    const int b0   = tile * 2;

    // Row assignment: lane L -> row r = L&15 of Z; halves split K within group of 4.
    const int r    = lane & 15;
    const int half = lane >> 4;          // 0: k%8 in {0..3}, 1: k%8 in {4..7}
    const int rv   = r & 7;              // (s*2+v) plane index
    const int bb   = b0 + (r >> 3);      // which batch element of the pair
    const float* rowp = views + (size_t)rv * (size_t)(B_ * D_)
                              + (size_t)bb * D_ + 4 * half;

    v8f c0 = {};                         // independent f32 accumulators
    v8f c1 = {};
    #pragma unroll 8
    for (int t = 0; t < 64; ++t) {       // 64 * 8 k-values = 512
        float4 q = *(const float4*)(rowp + 8 * t);   // global_load_b128, 16B aligned
        v2f a0 = { q.x, q.y };
        v2f a1 = { q.z, q.w };
        // same registers as A and B => Z * Z^T accumulation
        c0 = __builtin_amdgcn_wmma_f32_16x16x4_f32(false, a0, false, a0,
                                                   (short)0, c0, false, false);
        c1 = __builtin_amdgcn_wmma_f32_16x16x4_f32(false, a1, false, a1,
                                                   (short)0, c1, false, false);
    }

    // C/D layout: VGPR i holds (M = half*8 + i, N = lane&15)
    float* g = &gtile[wave][0];
    #pragma unroll
    for (int i = 0; i < 8; ++i)
        g[(half * 8 + i) * 16 + r] = c0[i] + c1[i];
    __syncthreads();

    // Epilogue: 8 lanes per wave -> one (batch, sample i) loss each.
    float lsum = 0.0f;
    if (lane < 8) {
        const int batch = lane >> 2;          // 0/1 within the pair
        const int i     = lane & 3;           // sample index
        const int base  = batch * 8;

        // inverse norms from the Gram diagonal; guard like the reference
        float invn[8];
        #pragma unroll
        for (int m = 0; m < 8; ++m) {
            float nrm = sqrtf(g[(base + m) * 16 + (base + m)]);
            invn[m] = 1.0f / fmaxf(nrm, 1e-12f);
        }

        const int   ri    = base + 2 * i;          // row of pos0_i
        const float invri = invn[2 * i];

        float vals[7];
        // positive: dot(z[i,0], z[i,1])
        vals[0] = g[ri * 16 + (ri + 1)] * invri * invn[2 * i + 1] * INV_TEMP;
        int nv = 1;
        #pragma unroll
        for (int j = 0; j < 4; ++j) {
            if (j == i) continue;                  // masked (-inf) entries dropped
            #pragma unroll
            for (int v = 0; v < 2; ++v) {
                const int cn = base + 2 * j + v;
                vals[nv++] = g[ri * 16 + cn] * invri * invn[2 * j + v] * INV_TEMP;
            }
        }
        float mx = vals[0];
        #pragma unroll
        for (int k = 1; k < 7; ++k) mx = fmaxf(mx, vals[k]);
        float se = 0.0f;
        #pragma unroll
        for (int k = 0; k < 7; ++k) se += expf(vals[k] - mx);
        lsum = (logf(se) + mx) - vals[0];          // logsumexp - positive logit
    }

    if (lane < 8) lred[wave][lane] = lsum;
    __syncthreads();
    if (lane == 0) {
        float s = 0.0f;
        #pragma unroll
        for (int k = 0; k < 8; ++k) s += lred[wave][k];
        wsum[wave] = s;
    }
    __syncthreads();
    if (tid == 0) {
        float s = 0.0f;
        #pragma unroll
        for (int w = 0; w < 8; ++w) s += wsum[w];
        partial[blockIdx.x] = s;
    }
}

// Deterministic final tree reduction -> mean
__global__ __launch_bounds__(256) void ntxent_finalize_kernel(
    const float* __restrict__ partial, float* __restrict__ out, int n)
{
    __shared__ float sh[256];
    float s = 0.0f;
    for (int i = threadIdx.x; i < n; i += 256) s += partial[i];
    sh[threadIdx.x] = s;
    __syncthreads();
    for (int step = 128; step > 0; step >>= 1) {
        if (threadIdx.x < step) sh[threadIdx.x] += sh[threadIdx.x + step];
        __syncthreads();
    }
    if (threadIdx.x == 0) out[0] = sh[0] * (1.0f / (float)(S_ * B_));
}

extern "C" void kernel_launch(void* const* d_in, const int* in_sizes, int n_in,
                              void* d_out, int out_size, void* d_ws, size_t ws_size,
                              hipStream_t stream)
{
    const float* views   = (const float*)d_in[0];
    float*       partial = (float*)d_ws;              // 512 floats of scratch
    (void)in_sizes; (void)n_in; (void)out_size; (void)ws_size;

    const int blocks = (B_ / 2) / 8;                  // 4096 tiles / 8 waves = 512
    ntxent_gram_kernel<<<blocks, 256, 0, stream>>>(views, partial);
    ntxent_finalize_kernel<<<1, 256, 0, stream>>>(partial, (float*)d_out, blocks);
}